// Classifier_62938450755768
// MI455X (gfx1250) — compile-verified
//
#include <hip/hip_runtime.h>

typedef __attribute__((ext_vector_type(2))) float v2f;
typedef __attribute__((ext_vector_type(8))) float v8f;

#define GD 128      // feature dim
#define NC 16       // classes
#define LDSTR 132   // padded LDS row stride in floats (132 mod 64 = 4 -> bank-conflict-free)

// Native fp32 atomic add (no CAS loop), no return -> STOREcnt path.
__device__ __forceinline__ void atomic_add_f32(float* p, float v) {
  asm volatile("global_atomic_add_f32 %0, %1, off" :: "v"(p), "v"(v) : "memory");
}

// 4 consecutive fp32 atomic adds sharing one 64-bit address pair.
__device__ __forceinline__ void atomic_add_f32x4(float* p, float4 v) {
  asm volatile(
      "global_atomic_add_f32 %0, %1, off\n\t"
      "global_atomic_add_f32 %0, %2, off offset:4\n\t"
      "global_atomic_add_f32 %0, %3, off offset:8\n\t"
      "global_atomic_add_f32 %0, %4, off offset:12"
      :: "v"(p), "v"(v.x), "v"(v.y), "v"(v.z), "v"(v.w) : "memory");
}

// ---------------- utility kernels ----------------

__global__ void k_zero_f4(float4* __restrict__ p, int n4) {
  int i = blockIdx.x * blockDim.x + threadIdx.x;
  if (i < n4) p[i] = make_float4(0.f, 0.f, 0.f, 0.f);
}

__global__ void k_degrees(const int* __restrict__ src, const int* __restrict__ dst,
                          float* __restrict__ deg_out, float* __restrict__ deg_in, int E) {
  int e = blockIdx.x * blockDim.x + threadIdx.x;
  if (e < E) {
    atomic_add_f32(&deg_out[src[e]], 1.0f);
    atomic_add_f32(&deg_in[dst[e]], 1.0f);
  }
}

__global__ void k_rsqrt(float* __restrict__ d, int n) {
  int i = blockIdx.x * blockDim.x + threadIdx.x;
  if (i < n) d[i] = rsqrtf(fmaxf(d[i], 1.0f));
}

// ---------------- GEMM: T = (X * rs[row]) @ W   (X: n x 128, W: 128 x 128) ----------------
// 256 threads = 8 waves; block computes a 16-row x 128-col panel; wave w owns cols [16w,16w+16)
__global__ void __launch_bounds__(256) k_gemm128(
    const float* __restrict__ X, const float* __restrict__ rs,
    const float* __restrict__ W, float* __restrict__ T) {
  __shared__ float As[16 * LDSTR];
  const int row0 = blockIdx.x << 4;
  const int tid = threadIdx.x;

  // cooperative stage of scaled A tile (16x128 fp32) into LDS
  for (int i = tid; i < 16 * GD; i += 256) {
    const int r = i >> 7, c = i & (GD - 1);
    As[r * LDSTR + c] = X[(row0 + r) * GD + c] * rs[row0 + r];
  }
  __syncthreads();

  const int wave = tid >> 5;
  const int lane = tid & 31;
  const int half = lane >> 4;           // 0: K pair {k,k+1}; 1: K pair {k+2,k+3}
  const int arow = lane & 15;           // A row (M) within tile
  const int col  = (wave << 4) + (lane & 15);  // B/D column (N)

  __builtin_prefetch(&W[col], 0, 1);    // global_prefetch_b8 of W panel start

  v8f acc = {};
#pragma unroll 4
  for (int k = 0; k < GD; k += 4) {
    const int ka = k + (half << 1);
    v2f a = *(const v2f*)&As[arow * LDSTR + ka];   // ds_load_b64
    v2f b;
    b.x = W[(ka + 0) * GD + col];
    b.y = W[(ka + 1) * GD + col];
    acc = __builtin_amdgcn_wmma_f32_16x16x4_f32(false, a, false, b,
                                                (short)0, acc, false, false);
  }
  // C/D layout: VGPR r -> lanes 0-15: M=r, lanes 16-31: M=r+8; N=lane%16
  const int orow = row0 + (half << 3);
#pragma unroll
  for (int r = 0; r < 8; ++r)
    T[(orow + r) * GD + col] = acc[r];
}

// ---------------- edge scatter: AGG[dst] += T[src], 1 wave per edge ----------------
// Edge index is wave-uniform (readfirstlane) so src/dst loads take the scalar (SMEM) path.
__global__ void __launch_bounds__(256) k_scatter(
    const float* __restrict__ T, const int* __restrict__ src,
    const int* __restrict__ dst, float* __restrict__ AGG, int E) {
  const int lane = threadIdx.x & 31;
  int e = blockIdx.x * 8 + (threadIdx.x >> 5);
  if (e >= E) return;                       // wave-uniform guard
  e = __builtin_amdgcn_readfirstlane(e);    // force SGPR -> s_load of indices
  const int s = src[e];
  const int d = dst[e];
  const int c = lane << 2;                  // lane*4 floats -> 512B per edge
  const float4 v = *(const float4*)&T[(size_t)s * GD + c];   // coalesced gather
  atomic_add_f32x4(&AGG[(size_t)d * GD + c], v);
}

// ---------------- H = relu(AGG * rs_in[row] + b) ----------------
__global__ void k_post(const float* __restrict__ AGG, const float* __restrict__ rs,
                       const float* __restrict__ b, float* __restrict__ H, int n) {
  const unsigned idx = blockIdx.x * 256u + threadIdx.x;   // n*32 threads
  const int node = idx >> 5;
  if (node >= n) return;
  const int c = (idx & 31) << 2;
  const float s = rs[node];
  float4 v = *(const float4*)&AGG[node * GD + c];
  const float4 bb = *(const float4*)&b[c];
  v.x = fmaxf(fmaf(v.x, s, bb.x), 0.f);
  v.y = fmaxf(fmaf(v.y, s, bb.y), 0.f);
  v.z = fmaxf(fmaf(v.z, s, bb.z), 0.f);
  v.w = fmaxf(fmaf(v.w, s, bb.w), 0.f);
  *(float4*)&H[node * GD + c] = v;
}

// ---------------- classifier: out = H @ Wc + bc   (128 -> 16, one 16-wide tile) ----------------
__global__ void __launch_bounds__(256) k_classifier(
    const float* __restrict__ H, const float* __restrict__ Wc,
    const float* __restrict__ bc, float* __restrict__ out, int ntiles) {
  const int wave = threadIdx.x >> 5;
  const int lane = threadIdx.x & 31;
  const int tile = blockIdx.x * 8 + wave;
  if (tile >= ntiles) return;          // wave-uniform guard: EXEC stays all-1s for live waves
  const int row0 = tile << 4;
  const int half = lane >> 4;
  const int arow = lane & 15;
  const int col  = lane & 15;

  v8f acc = {};
#pragma unroll 4
  for (int k = 0; k < GD; k += 4) {
    const int ka = k + (half << 1);
    v2f a = *(const v2f*)&H[(row0 + arow) * GD + ka];
    v2f b;
    b.x = Wc[(ka + 0) * NC + col];
    b.y = Wc[(ka + 1) * NC + col];
    acc = __builtin_amdgcn_wmma_f32_16x16x4_f32(false, a, false, b,
                                                (short)0, acc, false, false);
  }
  const float bias = bc[col];
  const int orow = row0 + (half << 3);
#pragma unroll
  for (int r = 0; r < 8; ++r)
    out[(orow + r) * NC + col] = acc[r] + bias;
}

// ---------------- driver ----------------

extern "C" void kernel_launch(void* const* d_in, const int* in_sizes, int n_in,
                              void* d_out, int out_size, void* d_ws, size_t ws_size,
                              hipStream_t stream) {
  const float* h   = (const float*)d_in[0];
  const int*   src = (const int*)d_in[1];
  const int*   dst = (const int*)d_in[2];
  const float* W1  = (const float*)d_in[3];
  const float* b1  = (const float*)d_in[4];
  const float* W2  = (const float*)d_in[5];
  const float* b2  = (const float*)d_in[6];
  const float* Wc  = (const float*)d_in[7];
  const float* bc  = (const float*)d_in[8];
  float* out = (float*)d_out;

  const int N = in_sizes[0] / GD;      // 100000
  const int E = in_sizes[1];           // 600000
  const int ntiles = N / 16;           // 6250 (N % 16 == 0)

  float* ws     = (float*)d_ws;
  float* rs_out = ws;                              // N floats
  float* rs_in  = ws + N;                          // N floats
  float* bufT   = ws + 2 * (size_t)N;              // N*128 floats
  float* bufA   = bufT + (size_t)N * GD;           // N*128 floats

  const dim3 B(256);
  const int nf4_big = (N * GD) / 4;
  const int nsblk = (E * 8 + (8 * 8 - 1)) / (8 * 8) * 8; // unused; keep simple below
  (void)nsblk;

  // degrees -> rsqrt norms (deg_out|deg_in contiguous)
  k_zero_f4<<<((2 * N / 4) + 255) / 256, B, 0, stream>>>((float4*)rs_out, 2 * N / 4);
  k_degrees<<<(E + 255) / 256, B, 0, stream>>>(src, dst, rs_out, rs_in, E);
  k_rsqrt<<<(2 * N + 255) / 256, B, 0, stream>>>(rs_out, 2 * N);

  const int scatter_blocks = (E + 7) / 8;  // 8 edges (waves) per 256-thread block

  // layer 1: t1 = (h * rs_out) @ W1 ; agg1[dst] += t1[src] ; h1 = relu(agg1*rs_in + b1)
  k_gemm128<<<ntiles, B, 0, stream>>>(h, rs_out, W1, bufT);
  k_zero_f4<<<(nf4_big + 255) / 256, B, 0, stream>>>((float4*)bufA, nf4_big);
  k_scatter<<<scatter_blocks, B, 0, stream>>>(bufT, src, dst, bufA, E);
  k_post<<<(N * 32 + 255) / 256, B, 0, stream>>>(bufA, rs_in, b1, bufT, N);   // bufT = h1

  // layer 2
  k_gemm128<<<ntiles, B, 0, stream>>>(bufT, rs_out, W2, bufA);                // bufA = t2
  k_zero_f4<<<(nf4_big + 255) / 256, B, 0, stream>>>((float4*)bufT, nf4_big);
  k_scatter<<<scatter_blocks, B, 0, stream>>>(bufA, src, dst, bufT, E);       // bufT = agg2
  k_post<<<(N * 32 + 255) / 256, B, 0, stream>>>(bufT, rs_in, b2, bufA, N);   // bufA = h2

  // classifier head
  k_classifier<<<(ntiles + 7) / 8, B, 0, stream>>>(bufA, Wc, bc, out, ntiles);
}